// CosFace_69295002354039
// MI455X (gfx1250) — compile-verified
//
#include <hip/hip_runtime.h>

// CosFace margin + scale:  out = s * (logits - m * onehot(label)), label==-1 -> no margin.
// BATCH=4096, NUM_CLASSES=50257, fp32.  Pure streaming op: ~1.65 GB traffic,
// roofline ~71us at 23.3 TB/s on MI455X.
//
// v2 changes vs v1:
//  - 32-bit unsigned indexing (823 MB buffer -> byte offsets < 2^31), killing the
//    64-bit s_mul_u64 / v_add_nc_u64 address chains seen in the v1 disassembly.
//  - 4x pumped grid-stride loop: 4 float4 NT loads in flight per thread.
//  - Keeps gfx1250 paths: global_load/store_b128 with TH=NT (stream >> 192MB L2),
//    global_prefetch_b8 one super-stride ahead (speculative, OOB-safe).

typedef float v4f __attribute__((ext_vector_type(4)));

#define SCALE_S 64.0f
#define MARGIN_M 0.35f

__global__ __launch_bounds__(256) void cosface_scale_stream(
    const v4f* __restrict__ in4, v4f* __restrict__ out4,
    const float* __restrict__ in_f, float* __restrict__ out_f,
    unsigned n4, unsigned n) {
  const unsigned stride = gridDim.x * blockDim.x;       // in float4 units
  unsigned i = blockIdx.x * blockDim.x + threadIdx.x;

  // Main 4x-pumped grid-stride loop: 4 x float4 (64 B) in flight per thread.
  for (; i + 3u * stride < n4; i += 4u * stride) {
    __builtin_prefetch(&in4[i + 4u * stride], 0, 0);    // gfx1250 global_prefetch_b8
    v4f a = __builtin_nontemporal_load(&in4[i]);
    v4f b = __builtin_nontemporal_load(&in4[i + stride]);
    v4f c = __builtin_nontemporal_load(&in4[i + 2u * stride]);
    v4f d = __builtin_nontemporal_load(&in4[i + 3u * stride]);
    a *= SCALE_S;
    b *= SCALE_S;
    c *= SCALE_S;
    d *= SCALE_S;
    __builtin_nontemporal_store(a, &out4[i]);
    __builtin_nontemporal_store(b, &out4[i + stride]);
    __builtin_nontemporal_store(c, &out4[i + 2u * stride]);
    __builtin_nontemporal_store(d, &out4[i + 3u * stride]);
  }
  // Remaining single-float4 iterations.
  for (; i < n4; i += stride) {
    v4f a = __builtin_nontemporal_load(&in4[i]);
    __builtin_nontemporal_store(a * SCALE_S, &out4[i]);
  }
  // Scalar tail (n % 4 elements) — first few threads of block 0.
  // (n is exactly divisible by 4 for this problem; kept for generality.)
  unsigned tail_start = n4 << 2;
  unsigned tail = n - tail_start;
  if (blockIdx.x == 0 && threadIdx.x < tail) {
    unsigned t = tail_start + threadIdx.x;
    out_f[t] = in_f[t] * SCALE_S;
  }
}

__global__ __launch_bounds__(256) void cosface_margin_fixup(
    const float* __restrict__ logits, const int* __restrict__ labels,
    float* __restrict__ out, int batch, unsigned num_classes) {
  int row = blockIdx.x * blockDim.x + threadIdx.x;
  if (row >= batch) return;
  int lab = labels[row];
  if (lab < 0) return;  // label == -1: no margin applied
  unsigned idx = (unsigned)row * num_classes + (unsigned)lab;  // < 2^28, fits easily
  out[idx] = (logits[idx] - MARGIN_M) * SCALE_S;
}

extern "C" void kernel_launch(void* const* d_in, const int* in_sizes, int n_in,
                              void* d_out, int out_size, void* d_ws, size_t ws_size,
                              hipStream_t stream) {
  const float* logits = (const float*)d_in[0];
  const int* labels = (const int*)d_in[1];
  float* out = (float*)d_out;

  const unsigned n = (unsigned)in_sizes[0];     // 4096 * 50257 = 205,852,672
  const int batch = in_sizes[1];                // 4096
  const unsigned num_classes = n / (unsigned)batch;
  const unsigned n4 = n >> 2;

  // 8192 blocks x 256 threads (wave32: 8 waves/block) -> ~2.1M threads,
  // ~6 quad-pumped iterations each; ample MLP to saturate 23.3 TB/s.
  const int threads = 256;
  const int blocks = 8192;
  cosface_scale_stream<<<blocks, threads, 0, stream>>>(
      (const v4f*)logits, (v4f*)out, logits, out, n4, n);

  cosface_margin_fixup<<<(batch + threads - 1) / threads, threads, 0, stream>>>(
      logits, labels, out, batch, num_classes);
}